// MPS_32332513804866
// MI455X (gfx1250) — compile-verified
//
#include <hip/hip_runtime.h>

typedef __attribute__((ext_vector_type(2))) float v2f;
typedef __attribute__((ext_vector_type(8))) float v8f;

// D = A(16x4 f32) * B(4x16 f32) + C(16x16 f32)
#define WMMA4(A, B, C) \
  __builtin_amdgcn_wmma_f32_16x16x4_f32(false, (A), false, (B), (short)0, (C), false, false)

// MPS contraction: 16 samples per wave as columns of a 16x16 complex state.
// - State lives permanently in WMMA C/D-layout registers; the fixed D->B row
//   permutation pi is folded into the (constant) A operands, so the 62-step
//   chain runs with zero cross-lane shuffles.
// - Spin selection is branchless: S' = Bt0*S + dBt*(S .* spin), with the mask
//   applied per-column (= per-lane, uniform across VGPRs) via v_cndmask.
// - Complex product uses the 3-multiply (Karatsuba) form: 6 real 16x16x16
//   matmuls = 24 K4-WMMAs per step instead of 32. The B-side sum Sr+Si is
//   carried across steps as Ssum' = P3 - 2*P2 (one FMA per VGPR).
__global__ __launch_bounds__(256) void mps_wmma_kernel(
    const int* __restrict__ x,
    const float* __restrict__ left_r,  const float* __restrict__ left_i,
    const float* __restrict__ bulk_r,  const float* __restrict__ bulk_i,
    const float* __restrict__ right_r, const float* __restrict__ right_i,
    float* __restrict__ out, int N)
{
  const int lane = threadIdx.x & 31;
  const int wave = (blockIdx.x * (int)blockDim.x + (int)threadIdx.x) >> 5;
  const int base = wave * 16;          // first sample (column) of this wave
  const int p    = lane & 15;          // row index within a lane-half / column id
  const int hi   = lane >> 4;          // which lane-half

  // ---- pack this column's 64 spins into a 64-bit mask (bit t = x[n, t]) ----
  const int* xp = x + (long)(base + p) * 64;
  unsigned long long mask = 0ull;
  #pragma unroll
  for (int q = 0; q < 16; ++q) {
    int4 v = ((const int4*)xp)[q];
    mask |= ((unsigned long long)(v.x & 1)) << (4*q + 0);
    mask |= ((unsigned long long)(v.y & 1)) << (4*q + 1);
    mask |= ((unsigned long long)(v.z & 1)) << (4*q + 2);
    mask |= ((unsigned long long)(v.w & 1)) << (4*q + 3);
  }

  // ---- A-operand constants: A[p,c] = bulk^T[p, pi(c)] = bulk[pi(c), p] ----
  // pi(4k+j+2h) = 2k+j+8h maps B-operand row index -> D-layout row index.
  // Karatsuba needs the sums (Ar+Ai) as extra constant operands.
  v2f A0r[4], A0i[4], A0s[4], dAr[4], dAi[4], dAs[4];
  #pragma unroll
  for (int k = 0; k < 4; ++k) {
    #pragma unroll
    for (int vv = 0; vv < 2; ++vv) {
      int c   = 4*k + vv + 2*hi;
      int pi  = ((c >> 2) << 1) | (c & 1) | (((c >> 1) & 1) << 3);
      int idx = pi * 16 + p;
      float b0r = bulk_r[idx],       b0i = bulk_i[idx];
      float b1r = bulk_r[256 + idx], b1i = bulk_i[256 + idx];
      A0r[k][vv] = b0r;
      A0i[k][vv] = b0i;
      A0s[k][vv] = b0r + b0i;
      dAr[k][vv] = b1r - b0r;
      dAi[k][vv] = b1i - b0i;
      dAs[k][vv] = (b1r - b0r) + (b1i - b0i);
    }
  }

  // ---- initial state: column n = left[x[n,0]] (D-layout: vgpr j = rows j, j+8)
  float Sr[8], Si[8], Ss[8];
  {
    int s0 = (int)(mask & 1ull);
    const float* lr = left_r + s0 * 16;
    const float* li = left_i + s0 * 16;
    #pragma unroll
    for (int j = 0; j < 8; ++j) {
      int row = j + hi * 8;
      Sr[j] = lr[row];
      Si[j] = li[row];
      Ss[j] = Sr[j] + Si[j];
    }
  }

  // ---- 62 bulk steps ----
  #pragma unroll 1
  for (int t = 1; t <= 62; ++t) {
    bool s = (mask >> t) & 1ull;   // per-column spin, uniform across VGPRs
    float Smr[8], Smi[8], Sms[8];
    #pragma unroll
    for (int j = 0; j < 8; ++j) {
      Smr[j] = s ? Sr[j] : 0.0f;
      Smi[j] = s ? Si[j] : 0.0f;
      Sms[j] = s ? Ss[j] : 0.0f;
    }

    // P1 = Mr*Sr, P2 = Mi*Si, P3 = (Mr+Mi)*(Sr+Si), with M = A0 + spin*dA
    v8f p1 = {}, p2 = {}, p3 = {};
    #pragma unroll
    for (int k = 0; k < 4; ++k) {
      v2f br  = { Sr[2*k],  Sr[2*k+1]  };
      v2f bi  = { Si[2*k],  Si[2*k+1]  };
      v2f bs  = { Ss[2*k],  Ss[2*k+1]  };
      v2f bmr = { Smr[2*k], Smr[2*k+1] };
      v2f bmi = { Smi[2*k], Smi[2*k+1] };
      v2f bms = { Sms[2*k], Sms[2*k+1] };
      p1 = WMMA4(A0r[k], br,  p1);
      p2 = WMMA4(A0i[k], bi,  p2);
      p3 = WMMA4(A0s[k], bs,  p3);
      p1 = WMMA4(dAr[k], bmr, p1);
      p2 = WMMA4(dAi[k], bmi, p2);
      p3 = WMMA4(dAs[k], bms, p3);
    }

    // S'r = P1 - P2 ; S'i = P3 - P1 - P2 ; Ssum' = P3 - 2*P2
    #pragma unroll
    for (int j = 0; j < 8; ++j) {
      float a = p1[j], b = p2[j], c = p3[j];
      Sr[j] = a - b;
      Si[j] = (c - a) - b;
      Ss[j] = __builtin_fmaf(-2.0f, b, c);
    }
  }

  // ---- final: a_n = sum_r S[r,n] * right[x[n,63]][r]  (complex dot) ----
  {
    int sL = (int)((mask >> 63) & 1ull);
    const float* rr = right_r + sL * 16;
    const float* ri = right_i + sL * 16;
    float accr = 0.0f, acci = 0.0f;
    #pragma unroll
    for (int j = 0; j < 8; ++j) {
      int row = j + hi * 8;
      float wr = rr[row], wi = ri[row];
      accr += Sr[j] * wr - Si[j] * wi;
      acci += Sr[j] * wi + Si[j] * wr;
    }
    // combine the two lane-halves (rows 0-7 + rows 8-15 of each column)
    accr += __shfl_xor(accr, 16, 32);
    acci += __shfl_xor(acci, 16, 32);
    if (lane < 16) {
      out[base + lane]     = accr;
      out[N + base + lane] = acci;
    }
  }
}

extern "C" void kernel_launch(void* const* d_in, const int* in_sizes, int n_in,
                              void* d_out, int out_size, void* d_ws, size_t ws_size,
                              hipStream_t stream) {
  const int*   x       = (const int*)  d_in[0];
  const float* left_r  = (const float*)d_in[1];
  const float* left_i  = (const float*)d_in[2];
  const float* bulk_r  = (const float*)d_in[3];
  const float* bulk_i  = (const float*)d_in[4];
  const float* right_r = (const float*)d_in[5];
  const float* right_i = (const float*)d_in[6];
  float* out = (float*)d_out;

  const int N     = in_sizes[0] / 64;   // 131072
  const int waves = N / 16;             // 16 samples per wave
  const int block = 256;                // 8 waves per block
  const int grid  = (waves * 32) / block;

  mps_wmma_kernel<<<grid, block, 0, stream>>>(
      x, left_r, left_i, bulk_r, bulk_i, right_r, right_i, out, N);
}